// activation_sub_up_36842229465134
// MI455X (gfx1250) — compile-verified
//
#include <hip/hip_runtime.h>

// Streaming gated-add: out[:, :D] = pq[:, :D] + a * sigmoid(pq[:, D:])
//                      out[:, D:] = pq[:, D:]
// Pure bandwidth kernel: ~1.07 GB moved once => ~46 us floor at 23.3 TB/s.
// b128 NT loads/stores, single-base immediate-offset addressing (dim=1024
// specialization => row stride 8192 B fits the 24-bit IOFFSET), 8-row unroll
// with all 16 loads issued before any use for deep MLP. WMMA inapplicable.

#define TPB 256
#define RPB 8  // rows per block

// Native clang vector type: required by __builtin_nontemporal_load/store.
typedef float v4f __attribute__((ext_vector_type(4)));

__device__ __forceinline__ float fast_sigmoid(float x) {
  // 1/(1+exp(-x)): v_exp_f32 + v_rcp_f32
  return __builtin_amdgcn_rcpf(1.0f + __expf(-x));
}

__device__ __forceinline__ v4f gate4(v4f av, v4f p, v4f q) {
  v4f r;
  r.x = fmaf(av.x, fast_sigmoid(q.x), p.x);
  r.y = fmaf(av.y, fast_sigmoid(q.y), p.y);
  r.z = fmaf(av.z, fast_sigmoid(q.z), p.z);
  r.w = fmaf(av.w, fast_sigmoid(q.w), p.w);
  return r;
}

// ---------- Fast path: dim == 4*TPB (1024), rows % RPB == 0 ----------
// One block = 256 lanes covering the 256 v4f columns of a half-row, 8 rows.
// All 16 loads + 16 stores share one base address; offsets are immediates.
__global__ __launch_bounds__(TPB) void sigmoid_gate_add_1024(
    const float* __restrict__ pq, const float* __restrict__ a,
    float* __restrict__ out) {
  constexpr int QCOLS   = TPB;        // v4f columns per half-row (256)
  constexpr int RSTRIDE = 2 * QCOLS;  // v4f per full row (512 -> 8192 B)
  const int c = threadIdx.x;
  const long base = (long)blockIdx.x * (RPB * RSTRIDE) + c;

  const v4f* __restrict__ pq4  = reinterpret_cast<const v4f*>(pq);
  const v4f* __restrict__ a4p  = reinterpret_cast<const v4f*>(a);
  v4f* __restrict__       out4 = reinterpret_cast<v4f*>(out);

  __builtin_prefetch(&a4p[c], 0, 3);  // gfx1250 global_prefetch_b8
  const v4f av = a4p[c];              // RT-cached 4 KB broadcast operand

  v4f p[RPB], q[RPB];
#pragma unroll
  for (int i = 0; i < RPB; ++i) {
    p[i] = __builtin_nontemporal_load(&pq4[base + i * RSTRIDE]);
    q[i] = __builtin_nontemporal_load(&pq4[base + i * RSTRIDE + QCOLS]);
  }
#pragma unroll
  for (int i = 0; i < RPB; ++i) {
    const v4f r = gate4(av, p[i], q[i]);
    __builtin_nontemporal_store(r,    &out4[base + i * RSTRIDE]);
    __builtin_nontemporal_store(q[i], &out4[base + i * RSTRIDE + QCOLS]);
  }
}

// ---------- Generic fallback (any dim % 4 == 0, any rows) ----------
__global__ __launch_bounds__(TPB) void sigmoid_gate_add_generic(
    const float* __restrict__ pq, const float* __restrict__ a,
    float* __restrict__ out, int dim, int rows) {
  const int qcols = dim >> 2;
  const int row0  = (int)blockIdx.x * RPB;
  const long row_stride4 = (long)qcols * 2;

  const v4f* __restrict__ pq4  = reinterpret_cast<const v4f*>(pq);
  const v4f* __restrict__ a4p  = reinterpret_cast<const v4f*>(a);
  v4f* __restrict__       out4 = reinterpret_cast<v4f*>(out);

  const int rmax = (rows - row0 < RPB) ? (rows - row0) : RPB;
  for (int c = threadIdx.x; c < qcols; c += TPB) {
    const v4f av = a4p[c];
    for (int i = 0; i < rmax; ++i) {
      const long base = (long)(row0 + i) * row_stride4;
      v4f p = __builtin_nontemporal_load(&pq4[base + c]);
      v4f q = __builtin_nontemporal_load(&pq4[base + qcols + c]);
      const v4f r = gate4(av, p, q);
      __builtin_nontemporal_store(r, &out4[base + c]);
      __builtin_nontemporal_store(q, &out4[base + qcols + c]);
    }
  }
}

extern "C" void kernel_launch(void* const* d_in, const int* in_sizes, int n_in,
                              void* d_out, int out_size, void* d_ws, size_t ws_size,
                              hipStream_t stream) {
  const float* pq = (const float*)d_in[0];  // (B, 2*dim) fp32
  const float* a  = (const float*)d_in[1];  // (dim,)    fp32
  float* out = (float*)d_out;               // (B, 2*dim) fp32

  const int dim  = in_sizes[1];             // 1024
  const int rows = in_sizes[0] / (2 * dim); // 65536

  if (dim == 4 * TPB && (rows % RPB) == 0) {
    sigmoid_gate_add_1024<<<dim3(rows / RPB), dim3(TPB), 0, stream>>>(pq, a, out);
  } else {
    const int blocks = (rows + RPB - 1) / RPB;
    sigmoid_gate_add_generic<<<dim3(blocks), dim3(TPB), 0, stream>>>(pq, a, out, dim, rows);
  }
}